// IcoUpSample_8641474199781
// MI455X (gfx1250) — compile-verified
//
#include <hip/hip_runtime.h>

typedef __attribute__((ext_vector_type(2))) float v2f;
typedef __attribute__((ext_vector_type(8))) float v8f;

namespace {
constexpr int kVIn    = 40962;
constexpr int kVOut   = 163842;
constexpr int kCh     = 64;      // C_IN == C_OUT == 64
constexpr int kBatch  = 16;
constexpr int kVTiles = (kVOut + 15) / 16;   // 10241 (last tile has 2 valid verts)
constexpr int kTiles  = kBatch * kVTiles;    // 163856
}

// One wave computes a 64(C_OUT) x 16(vertex) fp32 output tile for one batch:
//   D = (0.5*W) (64x64, A-operand) x (x[i0]+x[i1]) (64x16, B-operand) + bias
// using 16 K-steps x 4 M-tiles of V_WMMA_F32_16X16X4_F32.
__global__ __launch_bounds__(256) void ico_up_wmma(
    const float* __restrict__ x,            // [B, 64, V_IN]
    const long long* __restrict__ up_idx,   // [V_OUT, 2] int64
    const float* __restrict__ W,            // [64, 64] row-major (o, c)
    const float* __restrict__ bias,         // [64]
    float* __restrict__ out)                // [B, 64, V_OUT]
{
  const int lane = threadIdx.x & 31;
  const int n    = lane & 15;   // tile column (vertex) / A-row M
  const int h    = lane >> 4;   // half-wave: selects K pair / row-half

  const int wavesPerBlock = blockDim.x >> 5;
  const int wave  = blockIdx.x * wavesPerBlock + (threadIdx.x >> 5);
  const int nwave = gridDim.x * wavesPerBlock;

  // ---- Preload W fragments (register resident), pre-scaled by 0.5 so the
  //      K=2 mean is free (no per-K-step multiply on the gather path).
  // A 16x4 f32 layout: lane m=lane&15; VGPR0 -> K = 2h, VGPR1 -> K = 2h+1.
  v2f wfrag[4][16];
  #pragma unroll
  for (int mt = 0; mt < 4; ++mt) {
    const float* wrow = W + (16 * mt + n) * kCh;
    #pragma unroll
    for (int k = 0; k < 16; ++k) {
      const int c = 4 * k + 2 * h;                       // even -> 8B aligned
      v2f w = *reinterpret_cast<const v2f*>(wrow + c);
      wfrag[mt][k].x = 0.5f * w.x;
      wfrag[mt][k].y = 0.5f * w.y;
    }
  }

  // ---- Bias fragment in C/D layout: VGPR r holds row M = r + 8h ----
  v8f bfrag[4];
  #pragma unroll
  for (int mt = 0; mt < 4; ++mt)
    bfrag[mt] = *reinterpret_cast<const v8f*>(bias + 16 * mt + 8 * h); // 32B aligned

  for (int t = wave; t < kTiles; t += nwave) {
    const int b  = t / kVTiles;
    const int vt = t - b * kVTiles;
    const int v  = vt * 16 + n;
    const bool valid = v < kVOut;
    const int vc = valid ? v : (kVOut - 1);   // clamp so EXEC stays all-1s for WMMA

    // coalesced 16B/lane read of both indices
    const int i0 = (int)up_idx[2 * (long long)vc + 0];
    const int i1 = (int)up_idx[2 * (long long)vc + 1];

    const float* xb = x + (size_t)b * kCh * kVIn;

    // acc initialized with bias (folds the +b for free)
    v8f acc[4];
    #pragma unroll
    for (int mt = 0; mt < 4; ++mt) acc[mt] = bfrag[mt];

    #pragma unroll
    for (int k = 0; k < 16; ++k) {
      // B 4x16 f32 layout: VGPR0 -> K-row (2h), VGPR1 -> K-row (2h+1), N = n.
      const int c0 = 4 * k + 2 * h;
      const float* r0 = xb + (size_t)c0 * kVIn;
      const float* r1 = r0 + kVIn;
      v2f bf;
      bf.x = r0[i0] + r0[i1];   // 2*interp[c0  ][v]  (0.5 folded into W)
      bf.y = r1[i0] + r1[i1];   // 2*interp[c0+1][v]
      #pragma unroll
      for (int mt = 0; mt < 4; ++mt) {
        acc[mt] = __builtin_amdgcn_wmma_f32_16x16x4_f32(
            /*neg_a=*/false, wfrag[mt][k],
            /*neg_b=*/false, bf,
            /*c_mod=*/(short)0, acc[mt],
            /*reuse_a=*/false, /*reuse_b=*/false);
      }
    }

    // D layout: VGPR r -> row M = r + 8h, column = n (lane) -> coalesced 64B bursts.
    // Non-temporal: output is write-once/streamed; keep L2 reserved for x.
    if (valid) {
      float* ob = out + (size_t)b * kCh * kVOut + v;
      #pragma unroll
      for (int mt = 0; mt < 4; ++mt) {
        #pragma unroll
        for (int r = 0; r < 8; ++r) {
          __builtin_nontemporal_store(acc[mt][r],
                                      ob + (size_t)(16 * mt + 8 * h + r) * kVOut);
        }
      }
    }
  }
}

extern "C" void kernel_launch(void* const* d_in, const int* in_sizes, int n_in,
                              void* d_out, int out_size, void* d_ws, size_t ws_size,
                              hipStream_t stream) {
  (void)in_sizes; (void)n_in; (void)out_size; (void)d_ws; (void)ws_size;
  const float*     x      = (const float*)d_in[0];
  const long long* up_idx = (const long long*)d_in[1];
  const float*     W      = (const float*)d_in[2];
  const float*     bias   = (const float*)d_in[3];
  float*           out    = (float*)d_out;

  dim3 block(256);   // 8 waves/WG
  dim3 grid(2048);   // 16384 waves, ~10 tiles each; wave-linear order keeps one
                     // batch's x slice hot in L2 across concurrent waves
  ico_up_wmma<<<grid, block, 0, stream>>>(x, up_idx, W, bias, out);
}